// SpatialCrossAttention_30150670418189
// MI455X (gfx1250) — compile-verified
//
#include <hip/hip_runtime.h>
#include <math.h>

typedef float v2f __attribute__((ext_vector_type(2)));
typedef float v8f __attribute__((ext_vector_type(8)));

#define WMMA_F32(a, b, c) \
  __builtin_amdgcn_wmma_f32_16x16x4_f32(false, (a), false, (b), (short)0, (c), false, false)

// ---------------------------------------------------------------------------
// GEMM 1: offattn[24576,192] = (queries + pos_emb) @ Wc^T + bc
// Wc = [W_off;W_attn] pre-concatenated. Wave: 32x64 strip (MT=2 x NT=4).
// K-chunked: stage 8 A-frags + 16 B-frags, then 32 WMMAs per chunk.
// ---------------------------------------------------------------------------
__global__ void k_gemm_offattn(const float* __restrict__ queries,
                               const float* __restrict__ pos_emb,
                               const float* __restrict__ Wc,   // (192,256)
                               const float* __restrict__ bc,   // (192)
                               float* __restrict__ out) {
  const int K = 256, NT = 4, NTN = 3; // 192 / 64
  int lane = threadIdx.x & 31;
  int gw = blockIdx.x * 8 + (threadIdx.x >> 5);
  int tm = (gw / NTN) * 32;
  int tn = (gw % NTN) * 64;
  int m = lane & 15, kh = lane >> 4;

  size_t aoff0 = (size_t)(tm + m) * K;
  size_t aoff1 = (size_t)(tm + 16 + m) * K;
  size_t poff0 = (size_t)((tm + m) & 4095) * K;
  size_t poff1 = (size_t)((tm + 16 + m) & 4095) * K;

  size_t woff[NT]; float bv[NT];
#pragma unroll
  for (int t = 0; t < NT; ++t) {
    int col = tn + t * 16 + m;
    woff[t] = (size_t)col * K;
    bv[t] = bc[col];
  }

  v8f acc[2][NT] = {};
  for (int k0 = 0; k0 < K; k0 += 16) {
    v2f A0[4], A1[4], B[4][NT];
#pragma unroll
    for (int s = 0; s < 4; ++s) {
      int k = k0 + s * 4 + kh * 2;
      v2f a0 = *(const v2f*)(queries + aoff0 + k);
      v2f p0 = *(const v2f*)(pos_emb + poff0 + k);
      v2f a1 = *(const v2f*)(queries + aoff1 + k);
      v2f p1 = *(const v2f*)(pos_emb + poff1 + k);
      A0[s] = a0 + p0;
      A1[s] = a1 + p1;
#pragma unroll
      for (int t = 0; t < NT; ++t) B[s][t] = *(const v2f*)(Wc + woff[t] + k);
    }
#pragma unroll
    for (int s = 0; s < 4; ++s)
#pragma unroll
      for (int t = 0; t < NT; ++t) {
        acc[0][t] = WMMA_F32(A0[s], B[s][t], acc[0][t]);
        acc[1][t] = WMMA_F32(A1[s], B[s][t], acc[1][t]);
      }
  }
#pragma unroll
  for (int mt = 0; mt < 2; ++mt) {
#pragma unroll
    for (int t = 0; t < NT; ++t) {
      int col = tn + t * 16 + m;
#pragma unroll
      for (int r = 0; r < 8; ++r) {
        int row = tm + mt * 16 + kh * 8 + r;
        out[(size_t)row * 192 + col] = acc[mt][t][r] + bv[t];
      }
    }
  }
}

// ---------------------------------------------------------------------------
// GEMM 2: value[33600,256] = (features(n,c,hw) + lvl_emb + cam_emb) @ W_val^T + b
// 5600 % 32 == 0 so a 32-row tile never crosses a camera boundary.
// ---------------------------------------------------------------------------
__global__ void k_gemm_value(const float* __restrict__ features,
                             const float* __restrict__ lvl_emb,
                             const float* __restrict__ cam_emb,
                             const float* __restrict__ W_val, const float* __restrict__ b_val,
                             float* __restrict__ out) {
  const int K = 256, NT = 4, NTN = 4, HW = 5600;
  int lane = threadIdx.x & 31;
  int gw = blockIdx.x * 8 + (threadIdx.x >> 5);
  int tm = (gw / NTN) * 32;
  int tn = (gw % NTN) * 64;
  int m = lane & 15, kh = lane >> 4;

  int cam = tm / HW; // uniform across the 32-row tile
  size_t foff0 = (size_t)cam * K * HW + ((tm + m) - cam * HW);
  size_t foff1 = foff0 + 16;
  size_t coff = (size_t)cam * K;

  size_t woff[NT]; float bv[NT];
#pragma unroll
  for (int t = 0; t < NT; ++t) {
    int col = tn + t * 16 + m;
    woff[t] = (size_t)col * K;
    bv[t] = b_val[col];
  }

  v8f acc[2][NT] = {};
  for (int k0 = 0; k0 < K; k0 += 8) {
    v2f A0[2], A1[2], B[2][NT];
#pragma unroll
    for (int s = 0; s < 2; ++s) {
      int k = k0 + s * 4 + kh * 2;
      float f00 = features[foff0 + (size_t)k * HW];
      float f01 = features[foff0 + (size_t)(k + 1) * HW];
      float f10 = features[foff1 + (size_t)k * HW];
      float f11 = features[foff1 + (size_t)(k + 1) * HW];
      v2f e = *(const v2f*)(lvl_emb + k);
      v2f ce = *(const v2f*)(cam_emb + coff + k);
      e += ce;
      v2f a0, a1;
      a0.x = f00 + e.x; a0.y = f01 + e.y;
      a1.x = f10 + e.x; a1.y = f11 + e.y;
      A0[s] = a0; A1[s] = a1;
#pragma unroll
      for (int t = 0; t < NT; ++t) B[s][t] = *(const v2f*)(W_val + woff[t] + k);
    }
#pragma unroll
    for (int s = 0; s < 2; ++s)
#pragma unroll
      for (int t = 0; t < NT; ++t) {
        acc[0][t] = WMMA_F32(A0[s], B[s][t], acc[0][t]);
        acc[1][t] = WMMA_F32(A1[s], B[s][t], acc[1][t]);
      }
  }
#pragma unroll
  for (int mt = 0; mt < 2; ++mt) {
#pragma unroll
    for (int t = 0; t < NT; ++t) {
      int col = tn + t * 16 + m;
#pragma unroll
      for (int r = 0; r < 8; ++r) {
        int row = tm + mt * 16 + kh * 8 + r;
        out[(size_t)row * 256 + col] = acc[mt][t][r] + bv[t];
      }
    }
  }
}

// ---------------------------------------------------------------------------
// Sampler: softmax over points, bilinear gather, attn-weighted sum,
// fused masked camera reduction: red[q,d] = (sum_n valid*out_n)/count
// One wave per (q, head); lane = dh channel.
// ---------------------------------------------------------------------------
__global__ void k_sample(const float* __restrict__ offattn, // (24576,192)
                         const float* __restrict__ refpts,  // (6,4096,4,2)
                         const float* __restrict__ value,   // (33600,256)
                         const int* __restrict__ bev_mask,  // (6,4096,4,2)
                         float* __restrict__ red,           // (4096,256)
                         float* __restrict__ betav) {       // (4096)
  const int Q = 4096, HW = 5600, Ws = 100, Hs = 56;
  int lane = threadIdx.x & 31;
  int wid = blockIdx.x * 8 + (threadIdx.x >> 5);
  int h = wid & 7;
  int q = wid >> 3;

  float acc = 0.f;
  float cnt = 0.f;
  for (int n = 0; n < 6; ++n) {
    int row = n * Q + q;
    int valid = 0;
#pragma unroll
    for (int dz = 0; dz < 4; ++dz) valid |= bev_mask[((size_t)row * 4 + dz) * 2];
    if (!valid) continue;
    cnt += 1.f;
    const float* oa = offattn + (size_t)row * 192;
    // softmax over 8 points for head h
    float aw[8];
    float mx = -1e30f;
#pragma unroll
    for (int p = 0; p < 8; ++p) { aw[p] = oa[128 + h * 8 + p]; mx = fmaxf(mx, aw[p]); }
    float s = 0.f;
#pragma unroll
    for (int p = 0; p < 8; ++p) { aw[p] = __expf(aw[p] - mx); s += aw[p]; }
    float inv = 1.f / s;

    size_t vb = (size_t)n * HW * 256 + h * 32 + lane;
    const float* rp = refpts + (size_t)row * 8;
    float vsum = 0.f;
#pragma unroll
    for (int p = 0; p < 8; ++p) {
      int dz = p & 3; // point index p = pp*4 + dz
      // loc*W - 0.5 == ref*W + raw_offset - 0.5 (offset normalize cancels)
      float x = rp[dz * 2 + 0] * (float)Ws + oa[(h * 8 + p) * 2 + 0] - 0.5f;
      float y = rp[dz * 2 + 1] * (float)Hs + oa[(h * 8 + p) * 2 + 1] - 0.5f;
      float xf = floorf(x), yf = floorf(y);
      int x0 = (int)xf, y0 = (int)yf;
      float wx1 = x - xf, wx0 = 1.f - wx1;
      float wy1 = y - yf, wy0 = 1.f - wy1;
      float sv = 0.f;
#pragma unroll
      for (int c = 0; c < 4; ++c) {
        int xi = x0 + (c & 1), yi = y0 + (c >> 1);
        float w = ((c & 1) ? wx1 : wx0) * ((c >> 1) ? wy1 : wy0);
        if (xi >= 0 && xi < Ws && yi >= 0 && yi < Hs)
          sv += w * value[vb + (size_t)(yi * Ws + xi) * 256];
      }
      vsum += aw[p] * sv;
    }
    acc += vsum * inv;
  }
  float c = fmaxf(cnt, 1.f);
  red[(size_t)q * 256 + h * 32 + lane] = acc / c;
  if (h == 0 && lane == 0) betav[q] = (cnt > 0.f) ? 1.f : 0.f;
}

// ---------------------------------------------------------------------------
// GEMM 3/4: out[4096,256] = A @ W^T + beta(row)*bias   (beta==null -> 1)
// ---------------------------------------------------------------------------
__global__ void k_gemm_std(const float* __restrict__ A,
                           const float* __restrict__ W,
                           const float* __restrict__ bvec,
                           const float* __restrict__ beta,
                           float* __restrict__ out) {
  const int K = 256, NT = 4, NTN = 4;
  int lane = threadIdx.x & 31;
  int gw = blockIdx.x * 8 + (threadIdx.x >> 5);
  int tm = (gw / NTN) * 32;
  int tn = (gw % NTN) * 64;
  int m = lane & 15, kh = lane >> 4;

  size_t aoff0 = (size_t)(tm + m) * K;
  size_t aoff1 = (size_t)(tm + 16 + m) * K;
  size_t woff[NT]; float bv[NT];
#pragma unroll
  for (int t = 0; t < NT; ++t) {
    int col = tn + t * 16 + m;
    woff[t] = (size_t)col * K;
    bv[t] = bvec[col];
  }

  v8f acc[2][NT] = {};
  for (int k0 = 0; k0 < K; k0 += 16) {
    v2f A0[4], A1[4], B[4][NT];
#pragma unroll
    for (int s = 0; s < 4; ++s) {
      int k = k0 + s * 4 + kh * 2;
      A0[s] = *(const v2f*)(A + aoff0 + k);
      A1[s] = *(const v2f*)(A + aoff1 + k);
#pragma unroll
      for (int t = 0; t < NT; ++t) B[s][t] = *(const v2f*)(W + woff[t] + k);
    }
#pragma unroll
    for (int s = 0; s < 4; ++s)
#pragma unroll
      for (int t = 0; t < NT; ++t) {
        acc[0][t] = WMMA_F32(A0[s], B[s][t], acc[0][t]);
        acc[1][t] = WMMA_F32(A1[s], B[s][t], acc[1][t]);
      }
  }
#pragma unroll
  for (int mt = 0; mt < 2; ++mt) {
#pragma unroll
    for (int t = 0; t < NT; ++t) {
      int col = tn + t * 16 + m;
#pragma unroll
      for (int r = 0; r < 8; ++r) {
        int row = tm + mt * 16 + kh * 8 + r;
        float sc = beta ? beta[row] : 1.f;
        out[(size_t)row * 256 + col] = acc[mt][t][r] + sc * bv[t];
      }
    }
  }
}

// ---------------------------------------------------------------------------
extern "C" void kernel_launch(void* const* d_in, const int* in_sizes, int n_in,
                              void* d_out, int out_size, void* d_ws, size_t ws_size,
                              hipStream_t stream) {
  const float* queries  = (const float*)d_in[0];
  const float* pos_emb  = (const float*)d_in[1];
  const float* lvl_emb  = (const float*)d_in[2];
  const float* cam_emb  = (const float*)d_in[3];
  const float* features = (const float*)d_in[4];
  const float* refpts   = (const float*)d_in[5];
  const float* W_off    = (const float*)d_in[6];
  const float* b_off    = (const float*)d_in[7];
  const float* W_attn   = (const float*)d_in[8];
  const float* b_attn   = (const float*)d_in[9];
  const float* W_val    = (const float*)d_in[10];
  const float* b_val    = (const float*)d_in[11];
  const float* W_ao     = (const float*)d_in[12];
  const float* b_ao     = (const float*)d_in[13];
  const float* W_out    = (const float*)d_in[14];
  const float* b_out    = (const float*)d_in[15];
  const int*   bev_mask = (const int*)d_in[16];

  char* ws = (char*)d_ws;
  size_t o = 0;
  float* offattn = (float*)(ws + o); o += (size_t)24576 * 192 * 4;   // 18.9 MB
  float* value   = (float*)(ws + o); o += (size_t)33600 * 256 * 4;   // 34.4 MB
  float* red     = (float*)(ws + o); o += (size_t)4096 * 256 * 4;    //  4.2 MB
  float* slots   = (float*)(ws + o); o += (size_t)4096 * 256 * 4;    //  4.2 MB
  float* betav   = (float*)(ws + o); o += (size_t)4096 * 4;
  float* Wc      = (float*)(ws + o); o += (size_t)192 * 256 * 4;     // [W_off;W_attn]
  float* bcv     = (float*)(ws + o); o += (size_t)192 * 4;

  // Concatenate weights/biases so the GEMM sees one global base pointer.
  hipMemcpyAsync(Wc,              W_off,  (size_t)128 * 256 * 4, hipMemcpyDeviceToDevice, stream);
  hipMemcpyAsync(Wc + 128 * 256,  W_attn, (size_t)64 * 256 * 4,  hipMemcpyDeviceToDevice, stream);
  hipMemcpyAsync(bcv,             b_off,  (size_t)128 * 4,       hipMemcpyDeviceToDevice, stream);
  hipMemcpyAsync(bcv + 128,       b_attn, (size_t)64 * 4,        hipMemcpyDeviceToDevice, stream);

  // 24576/32 * 192/64 = 2304 waves -> 288 blocks of 8 waves
  k_gemm_offattn<<<288, 256, 0, stream>>>(queries, pos_emb, Wc, bcv, offattn);
  // 33600/32 * 256/64 = 4200 waves -> 525 blocks
  k_gemm_value<<<525, 256, 0, stream>>>(features, lvl_emb, cam_emb, W_val, b_val, value);
  // 4096 q * 8 heads = 32768 waves -> 4096 blocks
  k_sample<<<4096, 256, 0, stream>>>(offattn, refpts, value, bev_mask, red, betav);
  // 4096/32 * 256/64 = 512 waves -> 64 blocks
  k_gemm_std<<<64, 256, 0, stream>>>(red, W_ao, b_ao, betav, slots);
  k_gemm_std<<<64, 256, 0, stream>>>(slots, W_out, b_out, nullptr, (float*)d_out);
}